// SGATMultiLayer_75488345194755
// MI455X (gfx1250) — compile-verified
//
#include <hip/hip_runtime.h>
#include <hip/hip_bf16.h>
#include <math.h>

typedef float v2f __attribute__((ext_vector_type(2)));
typedef float v8f __attribute__((ext_vector_type(8)));

#define N_NODES 20000
#define N_EDGES 160000
#define IN_F    256
#define OUT_F   32
#define N_HEAD  4
#define BASES   4
#define SUPW    132           // (OUT_F+1)*N_HEAD
#define ASW     528           // (OUT_F+1)*BASES*N_HEAD
#define PRE_ALPHA  0.1f
#define POST_ALPHA 0.1f
#define EPSV 1e-9f

// ---------------------------------------------------------------------------
// K1: support0 = x @ weight via V_WMMA_F32_16X16X4_F32.
// One wave per 16x16 output tile; 8 waves per block; 156 blocks x 8 waves
// = 1248 of the (20000/16) x (128/16) = 10000 tiles. Output stride SUPW=132 so the GEMM
// result lands directly in the support[:, 0:32, :] slots.
// ---------------------------------------------------------------------------
__global__ __launch_bounds__(256) void gemm_wmma_kernel(
    const float* __restrict__ x, const float* __restrict__ w,
    float* __restrict__ support)
{
    const int lane = threadIdx.x & 31;
    const int wave = threadIdx.x >> 5;
    const int tile = blockIdx.x * 8 + wave;
    const int tm   = tile >> 3;      // 0..155  (M tile)
    const int tn   = tile & 7;       // 0..7     (N tile)
    const int hlf  = lane >> 4;      // 0/1: K pair select
    const int lrow = lane & 15;      // row (A) / col (B,C)

    const float* xrow = x + (tm * 16 + lrow) * IN_F;   // A: row = lrow
    const float* wcol = w + tn * 16 + lrow;            // B: col = lrow

    v8f acc = {0.f, 0.f, 0.f, 0.f, 0.f, 0.f, 0.f, 0.f};

    #pragma unroll 4
    for (int k0 = 0; k0 < IN_F; k0 += 4) {
        const int k = k0 + 2 * hlf;
        v2f a; a.x = xrow[k];               a.y = xrow[k + 1];
        v2f b; b.x = wcol[k * 128];         b.y = wcol[(k + 1) * 128];
        acc = __builtin_amdgcn_wmma_f32_16x16x4_f32(
            /*neg_a=*/false, a, /*neg_b=*/false, b,
            /*c_mod=*/(short)0, acc, /*reuse_a=*/false, /*reuse_b=*/false);
    }

    // C layout: VGPR r -> M = r + 8*hlf, N = lrow
    float* out = support + (tm * 16 + hlf * 8) * SUPW + tn * 16 + lrow;
    #pragma unroll
    for (int r = 0; r < 8; ++r)
        out[r * SUPW] = acc[r];
}

// K2: homogeneous coordinate support[:, 32, :] = 1
__global__ void fill_ones_kernel(float* __restrict__ support)
{
    int idx = blockIdx.x * blockDim.x + threadIdx.x;   // N*4
    if (idx >= N_NODES * N_HEAD) return;
    support[(idx >> 2) * SUPW + OUT_F * N_HEAD + (idx & 3)] = 1.0f;
}

// K3: attn1/attn2[a,k,j] = s + sqrt(s^2+1), s = sum_i support[a,i,j]*aX[i,k,j]
__global__ void attn_kernel(const float* __restrict__ support,
                            const float* __restrict__ a1,
                            const float* __restrict__ a2,
                            float* __restrict__ attn1,
                            float* __restrict__ attn2)
{
    int idx = blockIdx.x * blockDim.x + threadIdx.x;   // N*16
    if (idx >= N_NODES * BASES * N_HEAD) return;
    const int a  = idx >> 4;
    const int kj = idx & 15;              // k*4 + j
    const int j  = kj & 3;
    const float* srow = support + a * SUPW;
    float s1 = 0.f, s2 = 0.f;
    #pragma unroll 8
    for (int i = 0; i < OUT_F; ++i) {
        const float sv = srow[i * 4 + j];
        s1 += sv * a1[i * 16 + kj];
        s2 += sv * a2[i * 16 + kj];
    }
    attn1[idx] = s1 + sqrtf(s1 * s1 + 1.0f);
    attn2[idx] = s2 + sqrtf(s2 * s2 + 1.0f);
}

// K4: attn_support[a,i,k,j] = support[a,i,j] * attn2[a,k,j]
__global__ void outer_kernel(const float* __restrict__ support,
                             const float* __restrict__ attn2,
                             float* __restrict__ as)
{
    int idx = blockIdx.x * blockDim.x + threadIdx.x;   // N*528
    if (idx >= N_NODES * ASW) return;
    const int a = idx / ASW;
    const int r = idx - a * ASW;     // i*16 + k*4 + j
    const int i  = r >> 4;
    const int kj = r & 15;
    as[idx] = support[a * SUPW + i * 4 + (kj & 3)] * attn2[a * 16 + kj];
}

// K5: COO SpMM scatter: out[row] += v * in[col], feature width 528,
// one thread per (edge, 4-float chunk); GLOBAL_ATOMIC_ADD_F32 accumulation.
__global__ void spmm_scatter_kernel(const int* __restrict__ rows,
                                    const int* __restrict__ cols,
                                    const float* __restrict__ vals,
                                    const float* __restrict__ in,
                                    float* __restrict__ out)
{
    int idx = blockIdx.x * blockDim.x + threadIdx.x;   // E * (528/4)
    if (idx >= N_EDGES * (ASW / 4)) return;
    const int e = idx / (ASW / 4);
    const int c = idx - e * (ASW / 4);
    const int r  = rows[e];
    const int cl = cols[e];
    const float v = vals[e];
    const float4 src = *reinterpret_cast<const float4*>(in + cl * ASW + c * 4);
    float* dst = out + r * ASW + c * 4;
    __hip_atomic_fetch_add(dst + 0, v * src.x, __ATOMIC_RELAXED, __HIP_MEMORY_SCOPE_AGENT);
    __hip_atomic_fetch_add(dst + 1, v * src.y, __ATOMIC_RELAXED, __HIP_MEMORY_SCOPE_AGENT);
    __hip_atomic_fetch_add(dst + 2, v * src.z, __ATOMIC_RELAXED, __HIP_MEMORY_SCOPE_AGENT);
    __hip_atomic_fetch_add(dst + 3, v * src.w, __ATOMIC_RELAXED, __HIP_MEMORY_SCOPE_AGENT);
}

// K6: acc = PRE_ALPHA*old + (1-PRE_ALPHA)*acc   (in-place on acc)
__global__ void mix_kernel(const float* __restrict__ oldbuf,
                           float* __restrict__ acc)
{
    int idx = blockIdx.x * blockDim.x + threadIdx.x;   // N*528
    if (idx >= N_NODES * ASW) return;
    acc[idx] = PRE_ALPHA * oldbuf[idx] + (1.0f - PRE_ALPHA) * acc[idx];
}

// K7: contract over bases, POST mix with original support, renormalize by
// homogeneous coordinate. One thread per (node, head).
__global__ void contract_kernel(const float* __restrict__ as,
                                const float* __restrict__ attn1,
                                const float* __restrict__ support_init,
                                float* __restrict__ support)
{
    int idx = blockIdx.x * blockDim.x + threadIdx.x;   // N*4
    if (idx >= N_NODES * N_HEAD) return;
    const int a = idx >> 2;
    const int j = idx & 3;
    const float* arow = as + a * ASW;
    const float t0 = attn1[a * 16 + 0 * 4 + j];
    const float t1 = attn1[a * 16 + 1 * 4 + j];
    const float t2 = attn1[a * 16 + 2 * 4 + j];
    const float t3 = attn1[a * 16 + 3 * 4 + j];

    // homogeneous row (i = 32) first -> denominator
    float s32 = arow[32 * 16 + 0 + j] * t0 + arow[32 * 16 + 4 + j] * t1
              + arow[32 * 16 + 8 + j] * t2 + arow[32 * 16 + 12 + j] * t3;
    float tH = support_init[a * SUPW + OUT_F * 4 + j] * POST_ALPHA
             + s32 * (1.0f - POST_ALPHA);
    const float denom = tH + EPSV;

    for (int i = 0; i <= OUT_F; ++i) {
        const float* p = arow + i * 16;
        float s = p[0 + j] * t0 + p[4 + j] * t1 + p[8 + j] * t2 + p[12 + j] * t3;
        float t = support_init[a * SUPW + i * 4 + j] * POST_ALPHA
                + s * (1.0f - POST_ALPHA);
        support[a * SUPW + i * 4 + j] = t / denom;
    }
}

// K8: output = support[:, :-1, :] / (support[:, -1:, :] + eps) + bias_p
__global__ void output_kernel(const float* __restrict__ support,
                              const float* __restrict__ bias,
                              float* __restrict__ out)
{
    int idx = blockIdx.x * blockDim.x + threadIdx.x;   // N*128
    if (idx >= N_NODES * OUT_F * N_HEAD) return;
    const int a   = idx >> 7;
    const int col = idx & 127;                 // i*4 + j
    const float denom = support[a * SUPW + OUT_F * 4 + (col & 3)] + EPSV;
    out[idx] = support[a * SUPW + col] / denom + bias[col];
}

extern "C" void kernel_launch(void* const* d_in, const int* in_sizes, int n_in,
                              void* d_out, int out_size, void* d_ws, size_t ws_size,
                              hipStream_t stream)
{
    const float* x     = (const float*)d_in[0];                 // [N, 256]
    const int*   ei    = (const int*)  d_in[1];                 // [2, E]
    const float* vals  = (const float*)d_in[2];                 // [E]
    const float* w     = (const float*)d_in[3];                 // [256, 128]
    const float* bias  = (const float*)d_in[4];                 // [1, 32, 4]
    const float* a1    = (const float*)d_in[5];                 // [32, 4, 4]
    const float* a2    = (const float*)d_in[6];                 // [32, 4, 4]
    float*       out   = (float*)d_out;                        // [N, 128]

    const int* rows = ei;
    const int* cols = ei + N_EDGES;

    // Workspace carve-up (floats): ~150.4 MB total
    float* support      = (float*)d_ws;
    float* support_init = support      + (size_t)N_NODES * SUPW;
    float* attn1        = support_init + (size_t)N_NODES * SUPW;
    float* attn2        = attn1        + (size_t)N_NODES * 16;
    float* bufA         = attn2        + (size_t)N_NODES * 16;   // outer product / "old"
    float* bufB         = bufA         + (size_t)N_NODES * ASW;  // spmm ping
    float* bufC         = bufB         + (size_t)N_NODES * ASW;  // spmm pong

    const int TB = 256;
    const int nSup  = N_NODES * SUPW;
    const int nAS   = N_NODES * ASW;
    const int nAttn = N_NODES * BASES * N_HEAD;
    const int nHom  = N_NODES * N_HEAD;
    const int nEdgT = N_EDGES * (ASW / 4);
    const int nOut  = N_NODES * OUT_F * N_HEAD;

    // 1) GEMM (WMMA fp32) + homogeneous ones
    gemm_wmma_kernel<<<N_NODES / 16 / 8 * 8 / 8, TB, 0, stream>>>(x, w, support);
    // (grid = 156 blocks x 8 waves = 1248 tiles)
    fill_ones_kernel<<<(nHom + TB - 1) / TB, TB, 0, stream>>>(support);

    // 2) keep pristine copy (support_old in the reference)
    hipMemcpyAsync(support_init, support, (size_t)nSup * sizeof(float),
                   hipMemcpyDeviceToDevice, stream);

    // 3) attn1 / attn2 from the initial support
    attn_kernel<<<(nAttn + TB - 1) / TB, TB, 0, stream>>>(support_init, a1, a2, attn1, attn2);

    // 4) POST_ORDER = 2 outer iterations
    for (int it = 0; it < 2; ++it) {
        outer_kernel<<<(nAS + TB - 1) / TB, TB, 0, stream>>>(support, attn2, bufA);

        // PRE iteration 1: B = 0.1*A + 0.9*spmm(A)
        hipMemsetAsync(bufB, 0, (size_t)nAS * sizeof(float), stream);
        spmm_scatter_kernel<<<(nEdgT + TB - 1) / TB, TB, 0, stream>>>(rows, cols, vals, bufA, bufB);
        mix_kernel<<<(nAS + TB - 1) / TB, TB, 0, stream>>>(bufA, bufB);

        // PRE iteration 2: C = 0.1*A + 0.9*spmm(B)
        hipMemsetAsync(bufC, 0, (size_t)nAS * sizeof(float), stream);
        spmm_scatter_kernel<<<(nEdgT + TB - 1) / TB, TB, 0, stream>>>(rows, cols, vals, bufB, bufC);
        mix_kernel<<<(nAS + TB - 1) / TB, TB, 0, stream>>>(bufA, bufC);

        // contract over bases + POST mix + renormalize -> support
        contract_kernel<<<(nHom + TB - 1) / TB, TB, 0, stream>>>(bufC, attn1, support_init, support);
    }

    // 5) final output
    output_kernel<<<(nOut + TB - 1) / TB, TB, 0, stream>>>(support, bias, out);
}